// GCNModel_18073222381931
// MI455X (gfx1250) — compile-verified
//
#include <hip/hip_runtime.h>
#include <hip/hip_bf16.h>

#define N_NODES   50000
#define N_EDGES   500000
#define IN_FEATS  128
#define N_HIDDEN  128
#define N_CLASSES 64

typedef __attribute__((ext_vector_type(2))) float v2f;
typedef __attribute__((ext_vector_type(8))) float v8f;

// ---------------------------------------------------------------------------
// Zero-fill kernel (workspace / output init; graph-capture safe)
// ---------------------------------------------------------------------------
__global__ __launch_bounds__(256) void zero_kernel(float* __restrict__ p, int n) {
    int i = blockIdx.x * blockDim.x + threadIdx.x;
    if (i < n) p[i] = 0.0f;
}

// ---------------------------------------------------------------------------
// Dense GEMM via V_WMMA_F32_16X16X4_F32:  out[M,N] = (relu?)A[M,128] @ W[128,N] + b
// One wave per 16x16 output tile. 8 waves (256 thr) per block -> 128 rows/block.
// A-frag  (16x4 f32, 2 VGPR): lane l, half h=l>>4: A[m=l&15][k0+2h + {0,1}]
// B-frag  (4x16 f32, 2 VGPR): lane l: W[k0+2h + {0,1}][n = l&15]
// C/D     (16x16 f32, 8 VGPR): vgpr r -> row r + 8h, col l&15
// ---------------------------------------------------------------------------
template <int N, bool RELU_A>
__global__ __launch_bounds__(256)
void gemm_wmma_f32(const float* __restrict__ A, const float* __restrict__ W,
                   const float* __restrict__ bias, float* __restrict__ out, int M) {
    constexpr int K = 128;
    const int lane  = threadIdx.x & 31;
    const int wave  = threadIdx.x >> 5;
    const int mTile = blockIdx.x * 8 + wave;
    if (mTile * 16 >= M) return;               // wave-uniform exit: EXEC stays all-1s
    const int nTile = blockIdx.y;
    const int half  = lane >> 4;
    const int l16   = lane & 15;
    const int nCol  = nTile * 16 + l16;

    const float bv = bias[nCol];
    v8f c;
#pragma unroll
    for (int i = 0; i < 8; ++i) c[i] = bv;     // D = A*B + (bias-broadcast C)

    const float* aRow = A + (size_t)(mTile * 16 + l16) * K;

#pragma unroll 4
    for (int k0 = 0; k0 < K; k0 += 4) {
        const int kk = k0 + 2 * half;
        v2f a = *(const v2f*)(aRow + kk);       // global_load_b64
        if (RELU_A) {
            a.x = fmaxf(a.x, 0.0f);
            a.y = fmaxf(a.y, 0.0f);
        }
        v2f b;
        b.x = W[(size_t)(kk + 0) * N + nCol];
        b.y = W[(size_t)(kk + 1) * N + nCol];
        // 8 args: (neg_a, A, neg_b, B, c_mod, C, reuse_a, reuse_b)
        c = __builtin_amdgcn_wmma_f32_16x16x4_f32(false, a, false, b,
                                                  (short)0, c, false, false);
    }

    float* oBase = out + (size_t)(mTile * 16 + half * 8) * N + nCol;
#pragma unroll
    for (int r = 0; r < 8; ++r) oBase[(size_t)r * N] = c[r];
}

// ---------------------------------------------------------------------------
// Edge aggregation: out[dst[e]] += h[src[e]] * w[e]
// One wave per edge: 32 lanes cover FEATS floats (V = FEATS/32 per lane).
// Coalesced 512B/256B row reads; HW f32 atomics for the scatter-add.
// ---------------------------------------------------------------------------
template <int FEATS>
__global__ __launch_bounds__(256)
void aggregate_kernel(const float* __restrict__ h, const int* __restrict__ src,
                      const int* __restrict__ dst, const float* __restrict__ ew,
                      float* __restrict__ out, int nEdges) {
    const int lane = threadIdx.x & 31;
    const int e    = blockIdx.x * 8 + (threadIdx.x >> 5);
    if (e >= nEdges) return;
    const int   s = src[e];
    const int   d = dst[e];
    const float w = ew[e];

    constexpr int V = FEATS / 32;
    const float* hp = h   + (size_t)s * FEATS + lane * V;
    float*       op = out + (size_t)d * FEATS + lane * V;

    float vals[V];
#pragma unroll
    for (int i = 0; i < V; ++i) vals[i] = hp[i];   // merges to b128/b64 load
#pragma unroll
    for (int i = 0; i < V; ++i) unsafeAtomicAdd(op + i, vals[i] * w);
}

// ---------------------------------------------------------------------------
// Host-side launcher
// ---------------------------------------------------------------------------
extern "C" void kernel_launch(void* const* d_in, const int* in_sizes, int n_in,
                              void* d_out, int out_size, void* d_ws, size_t ws_size,
                              hipStream_t stream) {
    const float* x   = (const float*)d_in[0];
    const int*   src = (const int*)  d_in[1];
    const int*   dst = (const int*)  d_in[2];
    const float* w1  = (const float*)d_in[3];
    const float* w2  = (const float*)d_in[4];
    const float* w3  = (const float*)d_in[5];
    const float* W1  = (const float*)d_in[6];
    const float* b1  = (const float*)d_in[7];
    const float* W2  = (const float*)d_in[8];
    const float* b2  = (const float*)d_in[9];
    const float* W3  = (const float*)d_in[10];
    const float* b3  = (const float*)d_in[11];
    float* out = (float*)d_out;

    float* wsA = (float*)d_ws;                       // 50000*128 f32 (25.6 MB)
    float* wsB = wsA + (size_t)N_NODES * N_HIDDEN;   // 50000*128 f32 (25.6 MB)

    const int mTiles   = (N_NODES + 15) / 16;        // 3125
    const dim3 blk(256);
    const dim3 gemmGrid128((mTiles + 7) / 8, N_HIDDEN / 16);   // (391, 8)
    const dim3 gemmGrid64 ((mTiles + 7) / 8, N_CLASSES / 16);  // (391, 4)
    const dim3 aggGrid((N_EDGES + 7) / 8);                     // 62500

    const int nH  = N_NODES * N_HIDDEN;   // 6.4M
    const int nC  = N_NODES * N_CLASSES;  // 3.2M
    const dim3 zeroGridH((nH + 255) / 256);
    const dim3 zeroGridC((nC + 255) / 256);

    // ---- Layer 1: H1 = X @ W1 + b1 ; AGG1 = scatter(H1) ----
    gemm_wmma_f32<N_HIDDEN, false><<<gemmGrid128, blk, 0, stream>>>(x, W1, b1, wsA, N_NODES);
    zero_kernel<<<zeroGridH, blk, 0, stream>>>(wsB, nH);
    aggregate_kernel<N_HIDDEN><<<aggGrid, blk, 0, stream>>>(wsA, src, dst, w1, wsB, N_EDGES);

    // ---- Layer 2: H2 = relu(AGG1) @ W2 + b2 ; AGG2 = scatter(H2) ----
    gemm_wmma_f32<N_HIDDEN, true><<<gemmGrid128, blk, 0, stream>>>(wsB, W2, b2, wsA, N_NODES);
    zero_kernel<<<zeroGridH, blk, 0, stream>>>(wsB, nH);
    aggregate_kernel<N_HIDDEN><<<aggGrid, blk, 0, stream>>>(wsA, src, dst, w2, wsB, N_EDGES);

    // ---- Layer 3: H3 = relu(AGG2) @ W3 + b3 ; OUT = scatter(H3) ----
    gemm_wmma_f32<N_CLASSES, true><<<gemmGrid64, blk, 0, stream>>>(wsB, W3, b3, wsA, N_NODES);
    zero_kernel<<<zeroGridC, blk, 0, stream>>>(out, nC);
    aggregate_kernel<N_CLASSES><<<aggGrid, blk, 0, stream>>>(wsA, src, dst, w3, out, N_EDGES);
}